// SAGE_44659069944419
// MI455X (gfx1250) — compile-verified
//
#include <hip/hip_runtime.h>
#include <hip/hip_bf16.h>

// GraphSAGE (pool aggregator), 3 layers, on MI455X (gfx1250, wave32).
//
// Optimization strategy (see analysis): hoist the per-edge MLP to per-node
// (16x less compute), run the memory-bound edge max-scatter out of L2
// (y/pooled are 25.6 MB each, fit in the 192 MB L2), and keep everything in
// fp32 using V_WMMA_F32_16X16X4_F32 since the GEMMs are a tiny fraction of
// runtime and fp32 preserves reference precision.

typedef float v2f __attribute__((ext_vector_type(2)));
typedef float v8f __attribute__((ext_vector_type(8)));

#define NN   50000          // N_NODES
#define DIN  128            // all layer inputs are 128-wide
#define LSTR 132            // LDS row stride (floats): 132%64=4 -> conflict-free

// ---------------------------------------------------------------------------
// y = relu(x @ Wp + bp)     x:[NN,128] Wp:[128,128] bp:[128] y:[NN,128]
// Block = 256 threads (8 waves). Block tile: 16 rows x 128 cols.
// Wave w computes the 16x16 tile at cols [16w,16w+16) via 32 fp32 WMMAs.
// ---------------------------------------------------------------------------
__global__ __launch_bounds__(256)
void sage_node_pool(const float* __restrict__ x, const float* __restrict__ Wp,
                    const float* __restrict__ bp, float* __restrict__ y)
{
    __shared__ __align__(16) float As[16 * LSTR];
    const int tid  = threadIdx.x;
    const int row0 = blockIdx.x * 16;

    __builtin_prefetch(Wp, 0, 1);   // global_prefetch_b8

    // cooperative load of 16x128 fp32 tile (512 float4s)
    for (int u = tid; u < 16 * (DIN / 4); u += 256) {
        const int r = u >> 5, c4 = u & 31;
        const float4 v = ((const float4*)(x + (size_t)(row0 + r) * DIN))[c4];
        float* d = &As[r * LSTR + c4 * 4];
        d[0] = v.x; d[1] = v.y; d[2] = v.z; d[3] = v.w;
    }
    __syncthreads();

    const int wave = tid >> 5, lane = tid & 31;
    const int n = lane & 15, half = lane >> 4, n0 = wave * 16;

    v8f c = {};
    // A frag (16x4 f32): lane<16 holds row n, K={k,k+1}; lane>=16: K={k+2,k+3}
    const float* aA = &As[n * LSTR + 2 * half];
    // B frag (4x16 f32): mirrored K-split across wave halves
    const float* bA = Wp + (size_t)(2 * half) * DIN + n0 + n;
#pragma unroll
    for (int k = 0; k < DIN; k += 4) {
        v2f a = *(const v2f*)(aA + k);
        v2f b; b.x = bA[(size_t)k * DIN]; b.y = bA[(size_t)(k + 1) * DIN];
        c = __builtin_amdgcn_wmma_f32_16x16x4_f32(false, a, false, b,
                                                  (short)0, c, false, false);
    }

    const float bias = bp[n0 + n];
    // C/D layout: VGPR r -> (M = r + 8*half, N = n0+n)
#pragma unroll
    for (int r = 0; r < 8; ++r) {
        float v = c[r] + bias;
        v = v > 0.f ? v : 0.f;
        y[(size_t)(row0 + r + 8 * half) * DIN + n0 + n] = v;
    }
}

// ---------------------------------------------------------------------------
// pooled[dst] = max(pooled[dst], y[src])  -- one wave per edge, float4/lane.
// Values are post-ReLU (>=0) so uint-bit atomicMax == float max, and the
// zero-initialized pooled buffer reproduces segment_max + zero-degree->0.
// ---------------------------------------------------------------------------
__global__ __launch_bounds__(256)
void sage_edge_max(const int* __restrict__ esrc, const int* __restrict__ edst,
                   const float* __restrict__ y, unsigned int* __restrict__ pooled,
                   int n_edges)
{
    const int e = blockIdx.x * 8 + (threadIdx.x >> 5);
    if (e >= n_edges) return;
    const int lane = threadIdx.x & 31;
    const int s = esrc[e], d = edst[e];
    const float4 v = ((const float4*)(y + (size_t)s * DIN))[lane];
    unsigned int* pd = pooled + (size_t)d * DIN + lane * 4;
    if (v.x > 0.f) atomicMax(pd + 0, __float_as_uint(v.x));
    if (v.y > 0.f) atomicMax(pd + 1, __float_as_uint(v.y));
    if (v.z > 0.f) atomicMax(pd + 2, __float_as_uint(v.z));
    if (v.w > 0.f) atomicMax(pd + 3, __float_as_uint(v.w));
}

__global__ void zero_f4(float4* __restrict__ p, int n4)
{
    const int i = blockIdx.x * blockDim.x + threadIdx.x;
    if (i < n4) p[i] = make_float4(0.f, 0.f, 0.f, 0.f);
}

// ---------------------------------------------------------------------------
// out = l2norm_relu(hin @ Ws + pooled @ Wn + b)   DOUT in {128, 64}
// Block = 32*(DOUT/16) threads; tile = 16 rows x DOUT cols (full rows in
// block -> l2 norm fused here). Bias pre-loaded into the C accumulator.
// ---------------------------------------------------------------------------
template <int DOUT>
__global__ void sage_node_out(const float* __restrict__ hin,
                              const float* __restrict__ pooled,
                              const float* __restrict__ Ws,
                              const float* __restrict__ Wn,
                              const float* __restrict__ b,
                              float* __restrict__ out)
{
    constexpr int NW = DOUT / 16;
    constexpr int NT = 32 * NW;
    __shared__ __align__(16) float As[16 * LSTR];
    __shared__ __align__(16) float Ps[16 * LSTR];
    __shared__ __align__(16) float Ht[16 * DOUT];
    __shared__ float scl[16];

    const int tid  = threadIdx.x;
    const int row0 = blockIdx.x * 16;

    __builtin_prefetch(Ws, 0, 1);
    __builtin_prefetch(Wn, 0, 1);

    for (int u = tid; u < 16 * (DIN / 4); u += NT) {
        const int r = u >> 5, c4 = u & 31;
        const float4 va = ((const float4*)(hin    + (size_t)(row0 + r) * DIN))[c4];
        const float4 vp = ((const float4*)(pooled + (size_t)(row0 + r) * DIN))[c4];
        float* da = &As[r * LSTR + c4 * 4];
        float* dp = &Ps[r * LSTR + c4 * 4];
        da[0] = va.x; da[1] = va.y; da[2] = va.z; da[3] = va.w;
        dp[0] = vp.x; dp[1] = vp.y; dp[2] = vp.z; dp[3] = vp.w;
    }
    __syncthreads();

    const int wave = tid >> 5, lane = tid & 31;
    const int n = lane & 15, half = lane >> 4, n0 = wave * 16;

    v8f c;
    {
        const float bias = b[n0 + n];
#pragma unroll
        for (int r = 0; r < 8; ++r) c[r] = bias;
    }

    const float* aA = &As[n * LSTR + 2 * half];
    const float* aP = &Ps[n * LSTR + 2 * half];
    const float* bS = Ws + (size_t)(2 * half) * DOUT + n0 + n;
    const float* bN = Wn + (size_t)(2 * half) * DOUT + n0 + n;
#pragma unroll
    for (int k = 0; k < DIN; k += 4) {
        v2f a = *(const v2f*)(aA + k);
        v2f w; w.x = bS[(size_t)k * DOUT]; w.y = bS[(size_t)(k + 1) * DOUT];
        c = __builtin_amdgcn_wmma_f32_16x16x4_f32(false, a, false, w,
                                                  (short)0, c, false, false);
    }
#pragma unroll
    for (int k = 0; k < DIN; k += 4) {
        v2f a = *(const v2f*)(aP + k);
        v2f w; w.x = bN[(size_t)k * DOUT]; w.y = bN[(size_t)(k + 1) * DOUT];
        c = __builtin_amdgcn_wmma_f32_16x16x4_f32(false, a, false, w,
                                                  (short)0, c, false, false);
    }

    // stage tile to LDS for the fused row-wise l2 norm
#pragma unroll
    for (int r = 0; r < 8; ++r)
        Ht[(r + 8 * half) * DOUT + n0 + n] = c[r];
    __syncthreads();

    if (tid < 16) {
        float s = 0.f;
        const float* hr = &Ht[tid * DOUT];
#pragma unroll 8
        for (int j = 0; j < DOUT; ++j) s += hr[j] * hr[j];
        scl[tid] = 1.0f / fmaxf(sqrtf(s), 1e-12f);
    }
    __syncthreads();

    for (int idx = tid; idx < 16 * DOUT; idx += NT) {
        float v = Ht[idx] * scl[idx / DOUT];
        out[(size_t)row0 * DOUT + idx] = v > 0.f ? v : 0.f;   // tile rows are contiguous
    }
}

// ---------------------------------------------------------------------------
extern "C" void kernel_launch(void* const* d_in, const int* in_sizes, int n_in,
                              void* d_out, int out_size, void* d_ws, size_t ws_size,
                              hipStream_t stream)
{
    const float* x    = (const float*)d_in[0];
    const int*   esrc = (const int*)d_in[1];
    const int*   edst = (const int*)d_in[2];
    const float* Wp1 = (const float*)d_in[3],  *bp1 = (const float*)d_in[4];
    const float* Ws1 = (const float*)d_in[5],  *Wn1 = (const float*)d_in[6],  *b1 = (const float*)d_in[7];
    const float* Wp2 = (const float*)d_in[8],  *bp2 = (const float*)d_in[9];
    const float* Ws2 = (const float*)d_in[10], *Wn2 = (const float*)d_in[11], *b2 = (const float*)d_in[12];
    const float* Wp3 = (const float*)d_in[13], *bp3 = (const float*)d_in[14];
    const float* Ws3 = (const float*)d_in[15], *Wn3 = (const float*)d_in[16], *b3 = (const float*)d_in[17];
    float* out = (float*)d_out;

    float* buf0 = (float*)d_ws;                      // 25.6 MB each
    float* buf1 = buf0 + (size_t)NN * DIN;
    float* buf2 = buf1 + (size_t)NN * DIN;

    const int NE = in_sizes[1];
    const dim3 gN(NN / 16);                          // 3125 row tiles
    const dim3 gE((NE + 7) / 8);                     // 8 edges / 256-thread block
    const int  n4 = NN * DIN / 4;
    const dim3 gZ((n4 + 255) / 256);

    // ---- layer 1 (in = x) ----
    sage_node_pool<<<gN, 256, 0, stream>>>(x, Wp1, bp1, buf0);
    zero_f4<<<gZ, 256, 0, stream>>>((float4*)buf1, n4);
    sage_edge_max<<<gE, 256, 0, stream>>>(esrc, edst, buf0, (unsigned int*)buf1, NE);
    sage_node_out<128><<<gN, 256, 0, stream>>>(x, buf1, Ws1, Wn1, b1, buf2);

    // ---- layer 2 (in = buf2) ----
    sage_node_pool<<<gN, 256, 0, stream>>>(buf2, Wp2, bp2, buf0);
    zero_f4<<<gZ, 256, 0, stream>>>((float4*)buf1, n4);
    sage_edge_max<<<gE, 256, 0, stream>>>(esrc, edst, buf0, (unsigned int*)buf1, NE);
    sage_node_out<128><<<gN, 256, 0, stream>>>(buf2, buf1, Ws2, Wn2, b2, buf0);

    // ---- layer 3 (in = buf0, DOUT = 64, -> d_out) ----
    sage_node_pool<<<gN, 256, 0, stream>>>(buf0, Wp3, bp3, buf2);
    zero_f4<<<gZ, 256, 0, stream>>>((float4*)buf1, n4);
    sage_edge_max<<<gE, 256, 0, stream>>>(esrc, edst, buf2, (unsigned int*)buf1, NE);
    sage_node_out<64><<<gN, 128, 0, stream>>>(buf0, buf1, Ws3, Wn3, b3, out);
}